// BasicBlock_57363583205649
// MI455X (gfx1250) — compile-verified
//
#include <hip/hip_runtime.h>

// ---------------------------------------------------------------------------
// MI455X (gfx1250) implementation of the active-shift BasicBlock.
// Both 1x1 grouped convs are mapped onto V_WMMA_F32_16X16X4_F32 (wave32).
// Memory-bound workload (~190 MB HBM @ 23.3 TB/s ~ 8 us); f32 WMMA keeps the
// 2.47 GFLOP of conv work off the VALU so the kernel stays bandwidth-limited.
// ---------------------------------------------------------------------------

typedef __attribute__((ext_vector_type(2))) float v2f;
typedef __attribute__((ext_vector_type(8))) float v8f;

#define IMG_H 56
#define IMG_W 56
#define HW    3136            // 56*56
#define NCH   96
#define CHW   (NCH * HW)      // per-image plane stride
#define NIMG  32
#define NPOS  (NIMG * HW)     // 100352 flattened (n,h,w) columns
#define BN_EPS 1e-5f

#define W1_PITCH 97           // 96 + 1 pad -> conflict-free LDS column reads
#define W2_PITCH 33           // 32 + 1 pad

// ===========================================================================
// Kernel 1: fmap = conv1x1_g2( bnrelu( shuffle(x, prev) ) )
//   shuffled channel c: [0,48)=x[c], [48,96)=prev[c], [96,144)=x[c-48],
//                       [144,192)=prev[c-144]
// Each wave: one strip of 16 columns x 96 output channels (6 WMMA D tiles).
// ===========================================================================
__global__ __launch_bounds__(256)
void conv1_wmma_kernel(const float* __restrict__ x,
                       const float* __restrict__ prev,
                       const float* __restrict__ g1, const float* __restrict__ b1,
                       const float* __restrict__ m1, const float* __restrict__ vv1,
                       const float* __restrict__ w1,
                       float* __restrict__ fmap)
{
    __shared__ float lw[NCH * W1_PITCH];   // W1[96 out][96 k-in-group]
    __shared__ float s1[192], t1[192];     // folded BN1: a = relu(v*s + t)

    const int tid = threadIdx.x;
    for (int i = tid; i < 96 * 96; i += 256) {
        int r = i / 96, k = i - r * 96;
        lw[r * W1_PITCH + k] = w1[i];
    }
    if (tid < 192) {
        float s = g1[tid] * rsqrtf(vv1[tid] + BN_EPS);
        s1[tid] = s;
        t1[tid] = b1[tid] - m1[tid] * s;
    }
    __syncthreads();

    const int lane  = tid & 31;
    const int wave  = tid >> 5;
    const int strip = blockIdx.x * 8 + wave;
    const int pos0  = strip * 16;                 // strips never cross images
    const int n     = pos0 / HW;
    const int hw0   = pos0 - n * HW;
    const int col   = lane & 15;
    const int half  = lane >> 4;                  // K sub-pair selector

    const float* xb = x    + (size_t)n * CHW + hw0 + col;
    const float* pb = prev + (size_t)n * CHW + hw0 + col;

    // shuffled-channel activation fetch with folded BN1 + ReLU
    auto act = [&](int c) -> float {
        const float* src;
        int ch;
        if (c < 96) { src = (c < 48) ? xb : pb; ch = c; }
        else if (c < 144) { src = xb; ch = c - 48; }
        else { src = pb; ch = c - 144; }
        float v = src[(size_t)ch * HW];
        return fmaxf(fmaf(v, s1[c], t1[c]), 0.0f);
    };

    const v8f vzero = {0.f,0.f,0.f,0.f,0.f,0.f,0.f,0.f};
    v8f acc[6];
    #pragma unroll
    for (int i = 0; i < 6; ++i) acc[i] = vzero;

    #pragma unroll
    for (int g = 0; g < 2; ++g) {
        for (int kk = 0; kk < 96; kk += 4) {
            // B fragment (4 x 16): lane holds K = kk + 2*half + {0,1}, N = col
            int c0 = g * 96 + kk + half * 2;
            v2f bf;
            bf.x = act(c0);
            bf.y = act(c0 + 1);
            #pragma unroll
            for (int mt = 0; mt < 3; ++mt) {
                // A fragment (16 x 4): lane holds M = row, K = kk+2*half+{0,1}
                int row = g * 48 + mt * 16 + col;
                v2f af;
                af.x = lw[row * W1_PITCH + kk + half * 2];
                af.y = lw[row * W1_PITCH + kk + half * 2 + 1];
                acc[g * 3 + mt] = __builtin_amdgcn_wmma_f32_16x16x4_f32(
                    false, af, false, bf, (short)0, acc[g * 3 + mt],
                    false, false);
            }
        }
    }

    // D tile: VGPR r -> out-channel = base + r + 8*half, column = col
    float* ob = fmap + (size_t)n * CHW + hw0 + col;
    #pragma unroll
    for (int g = 0; g < 2; ++g)
        #pragma unroll
        for (int mt = 0; mt < 3; ++mt) {
            v8f d = acc[g * 3 + mt];
            int chb = g * 48 + mt * 16 + half * 8;
            #pragma unroll
            for (int r = 0; r < 8; ++r)
                ob[(size_t)(chb + r) * HW] = d[r];
        }
}

// ===========================================================================
// Kernel 2: out = conv1x1_g3( active_shift( bnrelu(fmap) ) ) + x
// fmap (38.5 MB) is L2-resident (192 MB L2) so the 4-corner bilinear gather
// is cheap. Per wave: 16 columns x 96 output channels, K=32 per group.
// ===========================================================================
__global__ __launch_bounds__(256)
void conv2_wmma_kernel(const float* __restrict__ x,
                       const float* __restrict__ fmap,
                       const float* __restrict__ g2, const float* __restrict__ b2,
                       const float* __restrict__ m2, const float* __restrict__ vv2,
                       const float* __restrict__ w2,
                       const float* __restrict__ shiftp,
                       float* __restrict__ out)
{
    __shared__ float lw[NCH * W2_PITCH];   // W2[96 out][32 k-in-group]
    __shared__ float s2[96], t2[96];
    __shared__ int   siy[96], six[96];
    __shared__ float cw00[96], cw01[96], cw10[96], cw11[96];

    const int tid = threadIdx.x;
    for (int i = tid; i < 96 * 32; i += 256) {
        int r = i / 32, k = i - r * 32;
        lw[r * W2_PITCH + k] = w2[i];
    }
    if (tid < 96) {
        float s = g2[tid] * rsqrtf(vv2[tid] + BN_EPS);
        s2[tid] = s;
        t2[tid] = b2[tid] - m2[tid] * s;
        float dy = shiftp[tid * 2 + 0];
        float dx = shiftp[tid * 2 + 1];
        float fy = floorf(dy), fx = floorf(dx);
        siy[tid] = (int)fy;
        six[tid] = (int)fx;
        float wy = dy - fy, wx = dx - fx;
        cw00[tid] = (1.f - wy) * (1.f - wx);
        cw01[tid] = (1.f - wy) * wx;
        cw10[tid] = wy * (1.f - wx);
        cw11[tid] = wy * wx;
    }
    __syncthreads();

    const int lane  = tid & 31;
    const int wave  = tid >> 5;
    const int strip = blockIdx.x * 8 + wave;
    const int col   = lane & 15;
    const int half  = lane >> 4;
    const int pos   = strip * 16 + col;
    const int n     = pos / HW;
    const int rhw   = pos - n * HW;
    const int h     = rhw / IMG_W;
    const int wc    = rhw - h * IMG_W;

    const float* fb = fmap + (size_t)n * CHW;

    // bilinear sample of relu(bn2(fmap[c])) at (h+dy_c, w+dx_c), zero-padded
    auto sample = [&](int c) -> float {
        int y0 = h + siy[c];
        int x0 = wc + six[c];
        const float* p = fb + (size_t)c * HW;
        float s = s2[c], t = t2[c];
        float v00 = 0.f, v01 = 0.f, v10 = 0.f, v11 = 0.f;
        bool x0ok = (x0 >= 0) & (x0 < IMG_W);
        bool x1ok = (x0 + 1 >= 0) & (x0 + 1 < IMG_W);
        if (y0 >= 0 && y0 < IMG_H) {
            const float* py = p + y0 * IMG_W;
            if (x0ok) v00 = fmaxf(fmaf(py[x0],     s, t), 0.f);
            if (x1ok) v01 = fmaxf(fmaf(py[x0 + 1], s, t), 0.f);
        }
        if (y0 + 1 >= 0 && y0 + 1 < IMG_H) {
            const float* py = p + (y0 + 1) * IMG_W;
            if (x0ok) v10 = fmaxf(fmaf(py[x0],     s, t), 0.f);
            if (x1ok) v11 = fmaxf(fmaf(py[x0 + 1], s, t), 0.f);
        }
        return cw00[c] * v00 + cw01[c] * v01 + cw10[c] * v10 + cw11[c] * v11;
    };

    const v8f vzero = {0.f,0.f,0.f,0.f,0.f,0.f,0.f,0.f};
    v8f acc[6];
    #pragma unroll
    for (int i = 0; i < 6; ++i) acc[i] = vzero;

    #pragma unroll
    for (int g = 0; g < 3; ++g) {
        #pragma unroll
        for (int kk = 0; kk < 32; kk += 4) {
            int c0 = g * 32 + kk + half * 2;
            v2f bf;
            bf.x = sample(c0);
            bf.y = sample(c0 + 1);
            #pragma unroll
            for (int mt = 0; mt < 2; ++mt) {
                int row = g * 32 + mt * 16 + col;
                v2f af;
                af.x = lw[row * W2_PITCH + kk + half * 2];
                af.y = lw[row * W2_PITCH + kk + half * 2 + 1];
                acc[g * 2 + mt] = __builtin_amdgcn_wmma_f32_16x16x4_f32(
                    false, af, false, bf, (short)0, acc[g * 2 + mt],
                    false, false);
            }
        }
    }

    // epilogue: out = D + x  (each lane owns its own column `pos`)
    const float* xp = x   + (size_t)n * CHW + rhw;
    float*       op = out + (size_t)n * CHW + rhw;
    #pragma unroll
    for (int g = 0; g < 3; ++g)
        #pragma unroll
        for (int mt = 0; mt < 2; ++mt) {
            v8f d = acc[g * 2 + mt];
            int chb = g * 32 + mt * 16 + half * 8;
            #pragma unroll
            for (int r = 0; r < 8; ++r)
                op[(size_t)(chb + r) * HW] = d[r] + xp[(size_t)(chb + r) * HW];
        }
}

// ===========================================================================
extern "C" void kernel_launch(void* const* d_in, const int* in_sizes, int n_in,
                              void* d_out, int out_size, void* d_ws, size_t ws_size,
                              hipStream_t stream) {
    const float* x     = (const float*)d_in[0];
    const float* prev  = (const float*)d_in[1];
    const float* g1    = (const float*)d_in[2];
    const float* b1    = (const float*)d_in[3];
    const float* m1    = (const float*)d_in[4];
    const float* v1    = (const float*)d_in[5];
    const float* g2    = (const float*)d_in[6];
    const float* b2    = (const float*)d_in[7];
    const float* m2    = (const float*)d_in[8];
    const float* v2    = (const float*)d_in[9];
    const float* w1    = (const float*)d_in[10];
    const float* w2    = (const float*)d_in[11];
    const float* shift = (const float*)d_in[12];

    float* out  = (float*)d_out;
    float* fmap = out + (size_t)NPOS * NCH;   // second tuple element, reused
                                              // as L2-resident intermediate

    const int strips = NPOS / 16;             // 6272
    const int blocks = strips / 8;            // 784 (8 waves/block)

    conv1_wmma_kernel<<<blocks, 256, 0, stream>>>(x, prev, g1, b1, m1, v1,
                                                  w1, fmap);
    conv2_wmma_kernel<<<blocks, 256, 0, stream>>>(x, fmap, g2, b2, m2, v2,
                                                  w2, shift, out);
}